// ModifiedMAML_32555852103868
// MI455X (gfx1250) — compile-verified
//
#include <hip/hip_runtime.h>
#include <hip/hip_bf16.h>

#define LEN_C 1024
#define LR_F  0.01f
#define GAMMA_F 0.999f

typedef float v2f __attribute__((ext_vector_type(2)));
typedef float v8f __attribute__((ext_vector_type(8)));

// ---------------------------------------------------------------------------
// Phase 1: anti[e], error[e], control_weights (to workspace), gamma vector.
// One block of 1024 threads (32 waves). Deterministic LDS tree reduction.
// ---------------------------------------------------------------------------
__global__ __launch_bounds__(1024) void maml_prep_kernel(
    const float* __restrict__ Fx,    // [16][4][1024]  (rs, e, n)
    const float* __restrict__ Dis,   // [4][1024]
    const float* __restrict__ w0,    // [16][1024]
    float* __restrict__ cw,          // workspace: [16][1024]
    float* __restrict__ out)         // d_out: [4096 anti_noise][1024 gamma]
{
    __shared__ float4 red[1024];
    __shared__ float  sErr[4];

    const int t = threadIdx.x;   // n index, 0..1023

    // anti[e] partial sums: thread t handles column n=t across all (r,s)
    float a0 = 0.f, a1 = 0.f, a2 = 0.f, a3 = 0.f;
    #pragma unroll
    for (int rs = 0; rs < 16; ++rs) {
        const float  w = w0[(rs << 10) + t];
        const float* f = Fx + ((rs << 2) << 10) + t;
        a0 += f[0]      * w;
        a1 += f[1024]   * w;
        a2 += f[2048]   * w;
        a3 += f[3072]   * w;
    }
    red[t] = make_float4(a0, a1, a2, a3);
    __syncthreads();

    // deterministic tree reduction over 1024 partials
    for (int s = 512; s > 0; s >>= 1) {
        if (t < s) {
            float4 x = red[t], y = red[t + s];
            red[t] = make_float4(x.x + y.x, x.y + y.y, x.z + y.z, x.w + y.w);
        }
        __syncthreads();
    }

    if (t < 4) {
        float4 anti4 = red[0];
        float anti = (t == 0) ? anti4.x : (t == 1) ? anti4.y : (t == 2) ? anti4.z : anti4.w;
        sErr[t] = Dis[(t << 10) + (LEN_C - 1)] - anti;
    }
    __syncthreads();

    const float e0 = sErr[0], e1 = sErr[1], e2 = sErr[2], e3 = sErr[3];

    // control_weights = w0 + LR * sum_e Fx[rs,e,t] * error[e]
    #pragma unroll
    for (int rs = 0; rs < 16; ++rs) {
        const float* f = Fx + ((rs << 2) << 10) + t;
        float s = f[0] * e0 + f[1024] * e1 + f[2048] * e2 + f[3072] * e3;
        cw[(rs << 10) + t] = w0[(rs << 10) + t] + LR_F * s;
    }

    // gamma_vector[t] = GAMMA^(1023 - t)
    out[4096 + t] = powf(GAMMA_F, (float)(1023 - t));
}

// ---------------------------------------------------------------------------
// Phase 2: causal Toeplitz mat-vec via fp32 WMMA (V_WMMA_F32_16X16X4_F32).
// Grid: (64 t-tiles, 4 e). One wave per block; EXEC all-ones at WMMA.
// A[i,k] = Fx[rs,e, t0+i-(n0+k)] (0 if negative), B[k,*] = cw[rs, n0+k]
// broadcast across N, C accumulates over all rs and K-blocks; any column of
// C is the 16 outputs for this t-tile.
// ---------------------------------------------------------------------------
__global__ __launch_bounds__(32) void maml_conv_wmma_kernel(
    const float* __restrict__ Fx,    // [16][4][1024]
    const float* __restrict__ cw,    // [16][1024]
    float* __restrict__ out)         // anti_noise [1024][4]
{
    const int tb   = blockIdx.x;         // 0..63 : 16-row tile of t
    const int e    = blockIdx.y;         // 0..3
    const int lane = threadIdx.x;        // 0..31
    const int i    = lane & 15;          // M within tile
    const int kh   = lane >> 4;          // K half (A/B layout: K = 2*kh + v)
    const int t0   = tb << 4;
    const int t    = t0 + i;
    const int nmax = t0 + 16;            // n in [0, nmax)

    v8f c = {0.f, 0.f, 0.f, 0.f, 0.f, 0.f, 0.f, 0.f};

    for (int rs = 0; rs < 16; ++rs) {
        const float* f = Fx + (((rs << 2) + e) << 10);
        const float* w = cw + (rs << 10);
        for (int n0 = 0; n0 < nmax; n0 += 4) {
            const int k    = n0 + (kh << 1);
            const int idx0 = t - k;          // <= 1023 always
            const int idx1 = idx0 - 1;
            v2f a, b;
            a.x = (idx0 >= 0) ? f[idx0] : 0.0f;   // causal mask
            a.y = (idx1 >= 0) ? f[idx1] : 0.0f;
            b.x = w[k];                            // broadcast across N
            b.y = w[k + 1];
            // (neg_a, A, neg_b, B, c_mod, C, reuse_a, reuse_b)
            c = __builtin_amdgcn_wmma_f32_16x16x4_f32(
                    false, a, false, b, (short)0, c, false, false);
        }
    }

    // C layout: VGPR r, lanes 0-15 -> M=r; lanes 16-31 -> M=8+r. Column N=0
    // sits in lanes 0 and 16; all columns are equal by construction.
    if ((lane & 15) == 0) {
        const int mb = t0 + (kh << 3);
        #pragma unroll
        for (int r = 0; r < 8; ++r)
            out[(mb + r) * 4 + e] = c[r];
    }
}

// ---------------------------------------------------------------------------
extern "C" void kernel_launch(void* const* d_in, const int* in_sizes, int n_in,
                              void* d_out, int out_size, void* d_ws, size_t ws_size,
                              hipStream_t stream) {
    const float* Fx  = (const float*)d_in[0];   // [4,4,4,1024]
    const float* Dis = (const float*)d_in[1];   // [4,1024]
    const float* w0  = (const float*)d_in[2];   // [4,4,1024]
    float* out = (float*)d_out;                 // 4096 anti_noise + 1024 gamma
    float* cw  = (float*)d_ws;                  // 64 KB control_weights scratch

    maml_prep_kernel<<<1, 1024, 0, stream>>>(Fx, Dis, w0, cw, out);

    dim3 grid(64, 4);
    maml_conv_wmma_kernel<<<grid, 32, 0, stream>>>(Fx, cw, out);
}